// anchor_50611894616702
// MI455X (gfx1250) — compile-verified
//
#include <hip/hip_runtime.h>
#include <hip/hip_bf16.h>
#include <math.h>

// Problem constants (from reference): B=32, A=3, C=14, G=152
#define GRID    152
#define GG      23104           // 152*152
#define NCH     42              // A * C = 3 * 14
#define NFEAT   15
#define BLK     256
#define PBPB    91              // ceil(23104 / 256); tail block has 64 valid positions
#define STRIDE  4.0f            // 608 / 152

typedef int v4i __attribute__((ext_vector_type(4)));
typedef __attribute__((address_space(1))) v4i* as1_v4i_p;
typedef __attribute__((address_space(3))) v4i* as3_v4i_p;

__device__ __forceinline__ float sigf(float v) {
    return 1.0f / (1.0f + __expf(-v));
}

__global__ __launch_bounds__(BLK) void decode_kernel(
    const float* __restrict__ x,        // (B, 42, 152, 152)
    const float* __restrict__ anchors,  // (3, 2)
    float* __restrict__ out)            // (B, 3*152*152, 15)
{
    // 3 per-anchor contiguous chunks of BLK*15 floats; chunk a maps 1:1 onto
    // the contiguous output region out[((b*3+a)*GG + p0)*15 ...].
    __shared__ __align__(16) float lds[3 * BLK * NFEAT];   // 46080 B

    const int tid = threadIdx.x;
    const int blk = blockIdx.x;        // position-block within batch
    const int b   = blockIdx.y;        // batch
    const int p0  = blk * BLK;
    const int p   = p0 + tid;          // linear grid position gy*152+gx
    const int count = (GG - p0 < BLK) ? (GG - p0) : BLK;

    if (p < GG) {
        const float* xb = x + (size_t)b * NCH * GG + p;   // channel c at xb[c*GG]
        const int gy = p / GRID;
        const int gx = p - gy * GRID;
        const float gxf = (float)gx;
        const float gyf = (float)gy;
        float* lrow = lds + tid * NFEAT;

        // Box features: f = 0..7, per anchor, from channels a*14 + {0..6}
        #pragma unroll
        for (int a = 0; a < 3; ++a) {
            const float* xc = xb + (size_t)(a * 14) * GG;
            float v0 = xc[(size_t)0 * GG];
            float v1 = xc[(size_t)1 * GG];
            float v2 = xc[(size_t)2 * GG];
            float v3 = xc[(size_t)3 * GG];
            float im = xc[(size_t)4 * GG];
            float re = xc[(size_t)5 * GG];
            float v6 = xc[(size_t)6 * GG];

            float ax   = floorf((sigf(v0) + gxf) * STRIDE);
            float ay   = floorf((sigf(v1) + gyf) * STRIDE);
            float aw   = __expf(v2) * anchors[a * 2 + 0];
            float ah   = __expf(v3) * anchors[a * 2 + 1];
            float conf = sigf(v6);
            float yaw  = atanf(im / re);

            float* l = lrow + a * (BLK * NFEAT);
            l[0] = im;  l[1] = re;  l[2] = yaw; l[3] = conf;
            l[4] = ax;  l[5] = ay;  l[6] = aw;  l[7] = ah;
        }

        // Class features: reference's reshape scrambles classes across anchors:
        // output (f, a_out), f=8+idx/3, a_out=idx%3  <-  channel (idx/7)*14 + 7 + idx%7
        #pragma unroll
        for (int idx = 0; idx < 21; ++idx) {
            const int a_src = idx / 7;
            const int k     = idx % 7;
            const int a_out = idx % 3;
            const int f     = 8 + idx / 3;
            float v = xb[(size_t)(a_src * 14 + 7 + k) * GG];
            lds[a_out * (BLK * NFEAT) + tid * NFEAT + f] = v;
        }
    }

    __syncthreads();

    // Drain LDS -> global: each anchor chunk is count*15 contiguous floats
    // (multiple of 4: 3840 or 960), 16B-aligned. Use CDNA5 async LDS->global
    // DMA stores when available.
    const int nvec = (count * NFEAT) >> 2;    // float4 count per chunk
    #pragma unroll
    for (int a = 0; a < 3; ++a) {
        float4* dst = (float4*)(out + ((size_t)(b * 3 + a) * GG + p0) * NFEAT);
        float4* src = (float4*)(lds + a * (BLK * NFEAT));
        for (int i = tid; i < nvec; i += BLK) {
#if defined(__AMDGCN__) && __has_builtin(__builtin_amdgcn_global_store_async_from_lds_b128)
            __builtin_amdgcn_global_store_async_from_lds_b128(
                (as1_v4i_p)(dst + i), (as3_v4i_p)(src + i), 0, 0);
#else
            dst[i] = src[i];
#endif
        }
    }

#if defined(__AMDGCN__) && __has_builtin(__builtin_amdgcn_s_wait_asynccnt)
    __builtin_amdgcn_s_wait_asynccnt(0);
#endif
}

extern "C" void kernel_launch(void* const* d_in, const int* in_sizes, int n_in,
                              void* d_out, int out_size, void* d_ws, size_t ws_size,
                              hipStream_t stream) {
    const float* x       = (const float*)d_in[0];
    const float* anchors = (const float*)d_in[1];
    float* out = (float*)d_out;

    const int B = in_sizes[0] / (NCH * GG);   // 32 for the reference setup
    dim3 grid(PBPB, B);
    decode_kernel<<<grid, BLK, 0, stream>>>(x, anchors, out);
}